// SphericalSegformer_10668698764013
// MI455X (gfx1250) — compile-verified
//
#include <hip/hip_runtime.h>
#include <hip/hip_bf16.h>
#include <math.h>

typedef float v2f __attribute__((ext_vector_type(2)));
typedef float v8f __attribute__((ext_vector_type(8)));

#define NB    4
#define CCH   256
#define HMID  64
#define WMID  128
#define HWMID (HMID*WMID)
#define HIN   128
#define WIN   256

// CDNA5 async global->LDS DMA (ASYNCcnt-tracked), probed at compile time.
// Probe-derived signature: (int4 as1* src, int4 as3* dst, imm offset, imm cpol)
#if __has_builtin(__builtin_amdgcn_global_load_async_to_lds_b128) && \
    __has_builtin(__builtin_amdgcn_s_wait_asynccnt)
#define HAVE_ASYNC_LDS 1
#else
#define HAVE_ASYNC_LDS 0
#endif

typedef int v4i_t __attribute__((vector_size(16)));
typedef __attribute__((address_space(1))) v4i_t gv4i_t;   // global int4
typedef __attribute__((address_space(3))) v4i_t lv4i_t;   // LDS int4

// LDS X-panel: 32 K rows x 128 N cols, row stride 136 floats.
// 136: (a) 544B rows keep every 16B chunk aligned for B128 DMA,
//      (b) 2*136 % 64 == 16 -> lanes 0-15 (rows r,r+1) and lanes 16-31
//          (rows r+2,r+3) hit disjoint bank groups on B-fragment reads.
#define XS_STRIDE 136

// ---------------------------------------------------------------------------
// 1x1-conv GEMM: Y[b,o,s] = bias[o] + res[b,o,s] + sum_c W[o,c] X[b,c,s]
// FP32 WMMA 16x16x4. Block = 256 threads (8 waves) -> 128(M) x 128(N) tile;
// each wave owns 16(M) x 128(N) via 8 accumulators, so one staged K-panel
// feeds 64 WMMAs per wave. CIN=256 gets unguarded async-B128 staging +
// float2 A-fragment loads; CIN=9 (pm combine) keeps the guarded path.
// ---------------------------------------------------------------------------
template <int CIN>
__global__ __launch_bounds__(256) void k_conv1x1_wmma(
    const float* __restrict__ X, const float* __restrict__ Wt,
    const float* __restrict__ bias, const float* res,
    float* Y, int Cout, int HW)
{
    __shared__ float Xs[32][XS_STRIDE];
    const int b    = blockIdx.z;
    const int n0   = blockIdx.x * 128;
    const int m0   = blockIdx.y * 128;
    const int t    = threadIdx.x;
    const int lane = t & 31;
    const int wave = t >> 5;                     // 0..7 -> M sub-tile
    const int l15  = lane & 15;
    const int hi2  = ((lane >> 4) & 1) * 2;      // K sub-offset: 0 or 2
    const int mrow = m0 + wave * 16 + l15;       // A-matrix row for this lane

    const v8f vzero = {0.f,0.f,0.f,0.f,0.f,0.f,0.f,0.f};
    v8f acc[8];
#pragma unroll
    for (int a = 0; a < 8; ++a) acc[a] = vzero;

    const float* Xb = X + (size_t)b * CIN * HW + n0;
    for (int k0 = 0; k0 < CIN; k0 += 32) {
        __syncthreads();
        if constexpr ((CIN & 127) == 0) {
            // Unguarded panel: 32x128 floats = 1024 16B chunks, 4 per thread.
#pragma unroll
            for (int u = 0; u < 4; ++u) {
                int q   = u * 256 + t;
                int row = q >> 5;
                int c4  = (q & 31) * 4;
                const float* gp = Xb + (size_t)(k0 + row) * HW + c4;
                float*       lp = &Xs[row][c4];
#if HAVE_ASYNC_LDS
                __builtin_amdgcn_global_load_async_to_lds_b128(
                    (gv4i_t*)gp, (lv4i_t*)lp, 0, 0);
#else
                *(float4*)lp = *(const float4*)gp;
#endif
            }
#if HAVE_ASYNC_LDS
            __builtin_amdgcn_s_wait_asynccnt(0);
#endif
        } else {
            // Guarded scalar panel with zero padding (CIN=9).
#pragma unroll
            for (int u = 0; u < 16; ++u) {
                int idx = u * 256 + t;
                int row = idx >> 7;
                int col = idx & 127;
                int kk  = k0 + row;
                Xs[row][col] = (kk < CIN) ? Xb[(size_t)kk * HW + col] : 0.f;
            }
        }
        __syncthreads();
        if (k0 + 32 < CIN)                       // prefetch next K panel
            __builtin_prefetch(Xb + (size_t)(k0 + 32) * HW + (t & 127), 0, 1);
#pragma unroll
        for (int kk = 0; kk < 32; kk += 4) {
            int kb = k0 + kk + hi2;
            v2f Af;
            if constexpr ((CIN & 3) == 0) {
                const float2 w2 = *(const float2*)(Wt + (size_t)mrow * CIN + kb);
                Af.x = w2.x; Af.y = w2.y;
            } else {
                Af.x = (kb     < CIN) ? Wt[(size_t)mrow * CIN + kb    ] : 0.f;
                Af.y = (kb + 1 < CIN) ? Wt[(size_t)mrow * CIN + kb + 1] : 0.f;
            }
#pragma unroll
            for (int a = 0; a < 8; ++a) {
                v2f Bf = {Xs[kk + hi2][a * 16 + l15],
                          Xs[kk + hi2 + 1][a * 16 + l15]};
                acc[a] = __builtin_amdgcn_wmma_f32_16x16x4_f32(
                    false, Af, false, Bf, (short)0, acc[a], false, false);
            }
        }
    }

    const int hioff = (lane >> 4) * 8;           // D-matrix: lanes 16-31 -> M+8
#pragma unroll
    for (int a = 0; a < 8; ++a) {
        union { v8f v; float f[8]; } u; u.v = acc[a];
#pragma unroll
        for (int r = 0; r < 8; ++r) {
            int o = m0 + wave * 16 + hioff + r;
            size_t oi = ((size_t)b * Cout + o) * HW + n0 + a * 16 + l15;
            float val = u.f[r];
            if (bias) val += bias[o];
            if (res)  val += res[oi];
            Y[oi] = val;
        }
    }
}

// ---------------------------------------------------------------------------
// pm disco, factored: t[b,(c,k),i,j] = sum_e pm_vals[k,i,e] * x[b,c,lat,lon]
// ---------------------------------------------------------------------------
__global__ __launch_bounds__(128) void k_pm_gather(
    const float* __restrict__ x, const int* __restrict__ lat,
    const int* __restrict__ lon, const float* __restrict__ vals,
    float* __restrict__ out)
{
    const int i = blockIdx.x, b = blockIdx.y, j = threadIdx.x;
    float acc[9];
#pragma unroll
    for (int q = 0; q < 9; ++q) acc[q] = 0.f;
    for (int e = 0; e < 160; ++e) {
        int la = lat[i * 160 + e];
        int lo = lon[(i * 160 + e) * WMID + j];
        float v0 = vals[(0 * HMID + i) * 160 + e];
        float v1 = vals[(1 * HMID + i) * 160 + e];
        float v2 = vals[(2 * HMID + i) * 160 + e];
#pragma unroll
        for (int c = 0; c < 3; ++c) {
            float xv = x[((size_t)(b * 3 + c)) * (HIN * WIN) + la * WIN + lo];
            acc[c * 3 + 0] += v0 * xv;
            acc[c * 3 + 1] += v1 * xv;
            acc[c * 3 + 2] += v2 * xv;
        }
    }
#pragma unroll
    for (int q = 0; q < 9; ++q)
        out[((size_t)b * 9 + q) * HWMID + i * WMID + j] = acc[q];
}

// ---------------------------------------------------------------------------
// Channel LayerNorm per (b,i,j); coalesced along j per channel iteration.
// ---------------------------------------------------------------------------
__global__ __launch_bounds__(128) void k_ln(
    const float* __restrict__ X, float* __restrict__ Y,
    const float* __restrict__ g, const float* __restrict__ bta)
{
    const int i = blockIdx.x, b = blockIdx.y, j = threadIdx.x;
    const size_t base = ((size_t)b * CCH) * HWMID + (size_t)i * WMID + j;
    float s = 0.f, ss = 0.f;
    for (int c = 0; c < CCH; ++c) {
        float v = X[base + (size_t)c * HWMID];
        s += v; ss += v * v;
    }
    float mean = s * (1.f / CCH);
    float var  = ss * (1.f / CCH) - mean * mean;
    float rstd = rsqrtf(var + 1e-5f);
    for (int c = 0; c < CCH; ++c) {
        float v = X[base + (size_t)c * HWMID];
        Y[base + (size_t)c * HWMID] = (v - mean) * rstd * g[c] + bta[c];
    }
}

// ---------------------------------------------------------------------------
// Neighborhood attention, flash-style online softmax over e=96 neighbors.
// One thread per (b, head, i, j); dh=32 q/o vectors in registers.
// bias: mask==False  <=>  qw==0  -> -1e9 (so att_mask never needs reading).
// ---------------------------------------------------------------------------
__global__ __launch_bounds__(128) void k_attn(
    const float* __restrict__ Q, const float* __restrict__ K,
    const float* __restrict__ V, float* __restrict__ O,
    const int* __restrict__ lat, const int* __restrict__ lon,
    const float* __restrict__ qw)
{
    const int i = blockIdx.x, hd = blockIdx.y, b = blockIdx.z, j = threadIdx.x;
    const size_t cb = ((size_t)b * CCH + hd * 32) * HWMID;
    const size_t sp = (size_t)i * WMID + j;
    float qr[32];
#pragma unroll
    for (int d = 0; d < 32; ++d) qr[d] = Q[cb + (size_t)d * HWMID + sp];
    const float scale = 0.17677669529663687f;    // 1/sqrt(32)
    float m = -INFINITY, l = 0.f;
    float o[32];
#pragma unroll
    for (int d = 0; d < 32; ++d) o[d] = 0.f;
    for (int e = 0; e < 96; ++e) {
        int la = lat[i * 96 + e];
        int lo = lon[(i * 96 + e) * WMID + j];
        size_t gp = (size_t)la * WMID + lo;
        float s = 0.f;
#pragma unroll
        for (int d = 0; d < 32; ++d) s += qr[d] * K[cb + (size_t)d * HWMID + gp];
        s *= scale;
        float qv = qw[i * 96 + e];
        s += (qv > 0.f) ? logf(fmaxf(qv, 1e-8f)) : -1e9f;
        float mn   = fmaxf(m, s);
        float corr = expf(m - mn);               // first iter: exp(-inf)=0
        float p    = expf(s - mn);
        l = l * corr + p;
#pragma unroll
        for (int d = 0; d < 32; ++d)
            o[d] = o[d] * corr + p * V[cb + (size_t)d * HWMID + gp];
        m = mn;
    }
    float inv = 1.f / l;
#pragma unroll
    for (int d = 0; d < 32; ++d) O[cb + (size_t)d * HWMID + sp] = o[d] * inv;
}

// ---------------------------------------------------------------------------
// Depthwise disco + exact GELU. Thread per (b,c,i,j), loop over 48 neighbors.
// ---------------------------------------------------------------------------
__global__ __launch_bounds__(128) void k_dw_gelu(
    const float* __restrict__ Z, float* __restrict__ O,
    const int* __restrict__ lat, const int* __restrict__ lon,
    const float* __restrict__ vals, const float* __restrict__ wk)
{
    const int i = blockIdx.x, c = blockIdx.y, b = blockIdx.z, j = threadIdx.x;
    const size_t cb = ((size_t)b * CCH + c) * HWMID;
    float a0 = 0.f, a1 = 0.f, a2 = 0.f;
    for (int e = 0; e < 48; ++e) {
        int la = lat[i * 48 + e];
        int lo = lon[(i * 48 + e) * WMID + j];
        float xv = Z[cb + (size_t)la * WMID + lo];
        a0 += vals[(0 * HMID + i) * 48 + e] * xv;
        a1 += vals[(1 * HMID + i) * 48 + e] * xv;
        a2 += vals[(2 * HMID + i) * 48 + e] * xv;
    }
    float s = wk[c * 3 + 0] * a0 + wk[c * 3 + 1] * a1 + wk[c * 3 + 2] * a2;
    float g = 0.5f * s * (1.f + erff(s * 0.7071067811865476f));
    O[cb + (size_t)i * WMID + j] = g;
}

// ---------------------------------------------------------------------------
// Head projection to 3 channels (tiny GEMM, VALU).
// ---------------------------------------------------------------------------
__global__ __launch_bounds__(128) void k_head(
    const float* __restrict__ X, float* __restrict__ Y,
    const float* __restrict__ w, const float* __restrict__ bi)
{
    const int i = blockIdx.x, b = blockIdx.y, j = threadIdx.x;
    const size_t sp = (size_t)i * WMID + j;
    float a0 = bi[0], a1 = bi[1], a2 = bi[2];
    for (int c = 0; c < CCH; ++c) {
        float v = X[((size_t)b * CCH + c) * HWMID + sp];
        a0 += w[0 * CCH + c] * v;
        a1 += w[1 * CCH + c] * v;
        a2 += w[2 * CCH + c] * v;
    }
    Y[((size_t)b * 3 + 0) * HWMID + sp] = a0;
    Y[((size_t)b * 3 + 1) * HWMID + sp] = a1;
    Y[((size_t)b * 3 + 2) * HWMID + sp] = a2;
}

// ---------------------------------------------------------------------------
// Bilinear upsample 64x128 -> 128x256 with precomputed indices/weights.
// ---------------------------------------------------------------------------
__global__ __launch_bounds__(256) void k_upsample(
    const float* __restrict__ X, float* __restrict__ Y,
    const int* __restrict__ lat_lo, const float* __restrict__ lat_w,
    const int* __restrict__ lon_lo, const float* __restrict__ lon_w)
{
    int idx = blockIdx.x * 256 + threadIdx.x;    // ((b*3+o)*128 + io)*256 + jo
    int jo = idx & 255;
    int io = (idx >> 8) & 127;
    int bo = idx >> 15;                          // b*3 + o
    int llo = lat_lo[io];
    int l1  = llo + 1; if (l1 > HMID - 1) l1 = HMID - 1;
    int jlo = lon_lo[jo];
    int j1  = (jlo + 1) & (WMID - 1);
    float lw = lat_w[io], jw = lon_w[jo];
    const float* Xb = X + (size_t)bo * HWMID;
    float v00 = Xb[llo * WMID + jlo], v01 = Xb[llo * WMID + j1];
    float v10 = Xb[l1  * WMID + jlo], v11 = Xb[l1  * WMID + j1];
    float x0 = v00 + (v10 - v00) * lw;
    float x1 = v01 + (v11 - v01) * lw;
    Y[idx] = x0 + (x1 - x0) * jw;
}

// ---------------------------------------------------------------------------
extern "C" void kernel_launch(void* const* d_in, const int* in_sizes, int n_in,
                              void* d_out, int out_size, void* d_ws, size_t ws_size,
                              hipStream_t stream)
{
    const float* x         = (const float*)d_in[0];
    const float* pm_w      = (const float*)d_in[1];
    const float* pm_ln_w   = (const float*)d_in[2];
    const float* pm_ln_b   = (const float*)d_in[3];
    const float* att_ln_w  = (const float*)d_in[4];
    const float* att_ln_b  = (const float*)d_in[5];
    const float* wq        = (const float*)d_in[6];
    const float* bq        = (const float*)d_in[7];
    const float* wk        = (const float*)d_in[8];
    const float* bk        = (const float*)d_in[9];
    const float* wv        = (const float*)d_in[10];
    const float* bv        = (const float*)d_in[11];
    const float* wo        = (const float*)d_in[12];
    const float* bo        = (const float*)d_in[13];
    const float* ffn_ln_w  = (const float*)d_in[14];
    const float* ffn_ln_b  = (const float*)d_in[15];
    const float* ffn_in_w  = (const float*)d_in[16];
    const float* ffn_in_b  = (const float*)d_in[17];
    const float* dw_w      = (const float*)d_in[18];
    const float* ffn_out_w = (const float*)d_in[19];
    const float* ffn_out_b = (const float*)d_in[20];
    const float* final_ln_w= (const float*)d_in[21];
    const float* final_ln_b= (const float*)d_in[22];
    const float* head_w    = (const float*)d_in[23];
    const float* head_b    = (const float*)d_in[24];
    const float* pm_vals   = (const float*)d_in[25];
    const float* dw_vals   = (const float*)d_in[26];
    const float* att_qw    = (const float*)d_in[27];
    const float* up_lat_w  = (const float*)d_in[28];
    const float* up_lon_w  = (const float*)d_in[29];
    const int*   pm_lat    = (const int*)d_in[30];
    const int*   pm_lon    = (const int*)d_in[31];
    const int*   dw_lat    = (const int*)d_in[32];
    const int*   dw_lon    = (const int*)d_in[33];
    const int*   att_lat   = (const int*)d_in[34];
    const int*   att_lon   = (const int*)d_in[35];
    // d_in[36] att_mask (bool) intentionally unused: qw==0 iff mask==False
    const int*   up_lat_lo = (const int*)d_in[37];
    const int*   up_lon_lo = (const int*)d_in[38];

    float* wsf = (float*)d_ws;
    const size_t ACT = (size_t)NB * CCH * HWMID;   // 8.39M floats / buffer
    float* hB  = wsf;            // persistent residual stream
    float* aB  = wsf + 1 * ACT;  // LN / attention output scratch
    float* qB  = wsf + 2 * ACT;
    float* kB  = wsf + 3 * ACT;
    float* vB  = wsf + 4 * ACT;
    float* tpm = wsf + 5 * ACT;                    // [B,9,HW]
    float* h3  = tpm + (size_t)NB * 9 * HWMID;     // [B,3,HW]
    float* outp = (float*)d_out;

    dim3 cgrid(HWMID / 128, CCH / 128, NB);
    dim3 pgrid(HMID, NB);

    // Prologue: disco_full = gather-reduce + K=9 GEMM, then LN
    k_pm_gather<<<pgrid, 128, 0, stream>>>(x, pm_lat, pm_lon, pm_vals, tpm);
    k_conv1x1_wmma<9><<<cgrid, 256, 0, stream>>>(tpm, pm_w, nullptr, nullptr, aB, CCH, HWMID);
    k_ln<<<pgrid, 128, 0, stream>>>(aB, hB, pm_ln_w, pm_ln_b);

    for (int r = 0; r < 2; ++r) {
        const size_t WOFF = (size_t)r * CCH * CCH;
        const size_t BOFF = (size_t)r * CCH;
        // attention block
        k_ln<<<pgrid, 128, 0, stream>>>(hB, aB, att_ln_w + BOFF, att_ln_b + BOFF);
        k_conv1x1_wmma<CCH><<<cgrid, 256, 0, stream>>>(aB, wq + WOFF, bq + BOFF, nullptr, qB, CCH, HWMID);
        k_conv1x1_wmma<CCH><<<cgrid, 256, 0, stream>>>(aB, wk + WOFF, bk + BOFF, nullptr, kB, CCH, HWMID);
        k_conv1x1_wmma<CCH><<<cgrid, 256, 0, stream>>>(aB, wv + WOFF, bv + BOFF, nullptr, vB, CCH, HWMID);
        k_attn<<<dim3(HMID, 8, NB), 128, 0, stream>>>(qB, kB, vB, aB, att_lat, att_lon, att_qw);
        k_conv1x1_wmma<CCH><<<cgrid, 256, 0, stream>>>(aB, wo + WOFF, bo + BOFF, hB, hB, CCH, HWMID);
        // FFN block
        k_ln<<<pgrid, 128, 0, stream>>>(hB, aB, ffn_ln_w + BOFF, ffn_ln_b + BOFF);
        k_conv1x1_wmma<CCH><<<cgrid, 256, 0, stream>>>(aB, ffn_in_w + WOFF, ffn_in_b + BOFF, nullptr, qB, CCH, HWMID);
        k_dw_gelu<<<dim3(HMID, CCH, NB), 128, 0, stream>>>(qB, kB, dw_lat, dw_lon, dw_vals, dw_w + (size_t)r * CCH * 3);
        k_conv1x1_wmma<CCH><<<cgrid, 256, 0, stream>>>(kB, ffn_out_w + WOFF, ffn_out_b + BOFF, hB, hB, CCH, HWMID);
    }

    // Epilogue: final LN, head, bilinear upsample
    k_ln<<<pgrid, 128, 0, stream>>>(hB, aB, final_ln_w, final_ln_b);
    k_head<<<pgrid, 128, 0, stream>>>(aB, h3, head_w, head_b);
    k_upsample<<<dim3((NB * 3 * HIN * WIN) / 256), 256, 0, stream>>>(
        h3, outp, up_lat_lo, up_lat_w, up_lon_lo, up_lon_w);

    (void)in_sizes; (void)n_in; (void)out_size; (void)ws_size;
}